// AttentionCopy_57732950393347
// MI455X (gfx1250) — compile-verified
//
#include <hip/hip_runtime.h>
#include <stdint.h>

// out[b,t,v] = sum_s attn[b,t,s] * (ids[b,s]==v)
// B=8, T=64, S=512, V=50000. Store-bandwidth bound (102.4 MB f32 output,
// HBM floor ~4.4us at 23.3 TB/s). One workgroup per (b,t,chunk): zero a
// 50KB vocab chunk in LDS, scatter via ds_add_f32 LDS atomics (range-
// filtered), then DMA the chunk to global with the CDNA5 Tensor Data Mover
// (tensor_store_from_lds). 50KB LDS/WG -> ~4 resident WGs per WGP so TDM
// drain overlaps other WGs' compute. Fallback = nontemporal b128 stores.

#define VOCAB  50000
#define BATCH  8
#define TDIM   64
#define SDIM   512
#define BLOCK  512
#define NCHUNK 4
#define CHUNK  (VOCAB / NCHUNK)        // 12500 floats = 50,000 bytes, /4 exact

typedef float f32x4 __attribute__((ext_vector_type(4)));

#if defined(__gfx1250__) && __has_builtin(__builtin_amdgcn_tensor_store_from_lds)
#define USE_TDM 1
typedef unsigned int u32x4 __attribute__((ext_vector_type(4)));
typedef int          i32x4 __attribute__((ext_vector_type(4)));
typedef int          i32x8 __attribute__((ext_vector_type(8)));
#endif

__global__ __launch_bounds__(BLOCK)
void AttnScatter_kernel(const int* __restrict__ ids,
                        const float* __restrict__ attn,
                        float* __restrict__ out)
{
    extern __shared__ float s_row[];   // CHUNK floats = 50,000 bytes
    const int t     = blockIdx.x;      // 0..63
    const int b     = blockIdx.y;      // 0..7
    const int chunk = blockIdx.z;      // 0..3
    const int tid   = threadIdx.x;     // 0..511
    const int lo    = chunk * CHUNK;   // vocab range [lo, lo+CHUNK)

    // ---- 1) zero the LDS chunk with wide ds stores (3125 x b128) ----
    f32x4* s4 = (f32x4*)s_row;
    const f32x4 z = { 0.f, 0.f, 0.f, 0.f };
    for (int i = tid; i < CHUNK / 4; i += BLOCK)
        s4[i] = z;
    __syncthreads();

    // ---- 2) scatter-add: one source position per thread (BLOCK==SDIM),
    //         keep only ids landing in this vocab chunk ----
    {
        const int   s  = tid;
        const int   id = ids[b * SDIM + s] - lo;
        const float a  = attn[((size_t)(b * TDIM + t)) * SDIM + s];
        if ((unsigned)id < (unsigned)CHUNK)
            atomicAdd(&s_row[id], a);  // ds_add_f32 (LDS atomic, no return)
    }
    __syncthreads();

    float* orow = out + (size_t)(b * TDIM + t) * VOCAB + lo;

#if USE_TDM
    // ---- 3a) TDM: one async DMA of the 50KB chunk, issued by wave 0 only.
    // readfirstlane forces a scalar branch (TDM ignores EXEC, so waves 1..15
    // must not reach the instruction).
    if (__builtin_amdgcn_readfirstlane((int)tid) == 0) {
        const uint64_t ga  = (uint64_t)(uintptr_t)orow;
        const uint32_t lds = (uint32_t)(uintptr_t)s_row;   // addr[31:0] = LDS byte addr

        // D# group 0 (128b): count=1 | lds_addr | global_addr[56:0] | type=2
        u32x4 g0;
        g0.x = 1u;                                          // count=1, user desc
        g0.y = lds;                                         // lds_addr (bytes)
        g0.z = (uint32_t)ga;                                // global_addr[31:0]
        g0.w = ((uint32_t)(ga >> 32) & 0x01FFFFFFu)         // global_addr[56:32]
             | 0x80000000u;                                 // type[127:126]=2 ("image")

        // D# group 1 (256b): 1-D tile, data_size=4B,
        // tensor_dim0 = tile_dim0 = dim0_stride = CHUNK, tensor_dim1 = tile_dim1 = 1.
        constexpr uint32_t D0 = (uint32_t)CHUNK;            // 12500
        i32x8 g1;
        g1[0] = (int)(2u << 16);                            // data_size=2 (4B), no flags
        g1[1] = (int)((D0 & 0xFFFFu) << 16);                // tensor_dim0[15:0] @ [63:48]
        g1[2] = (int)(((D0 >> 16) & 0xFFFFu) | (1u << 16)); // t_dim0[31:16] | tensor_dim1=1
        g1[3] = (int)((D0 & 0xFFFFu) << 16);                // tile_dim0 @ [127:112]
        g1[4] = 1;                                          // tile_dim1=1, tile_dim2=0
        g1[5] = (int)D0;                                    // tensor_dim0_stride[31:0]
        g1[6] = 0;                                          // stride hi / dim1_stride lo
        g1[7] = 0;

        i32x4 gz = { 0, 0, 0, 0 };   // groups 2/3 unused (1-D tile)

#if __clang_major__ >= 23
        i32x8 gz8 = { 0, 0, 0, 0, 0, 0, 0, 0 };
        __builtin_amdgcn_tensor_store_from_lds(g0, g1, gz, gz, gz8, 0);
#else
        __builtin_amdgcn_tensor_store_from_lds(g0, g1, gz, gz, 0);
#endif
#if __has_builtin(__builtin_amdgcn_s_wait_tensorcnt)
        __builtin_amdgcn_s_wait_tensorcnt(0);   // hold LDS until DMA drains
#endif
    }
#else
    // ---- 3b) fallback: streaming nontemporal b128 stores (write-once data) ----
    f32x4* o4 = (f32x4*)orow;
    for (int i = tid; i < CHUNK / 4; i += BLOCK)
        __builtin_nontemporal_store(s4[i], o4 + i);
#endif
}

extern "C" void kernel_launch(void* const* d_in, const int* in_sizes, int n_in,
                              void* d_out, int out_size, void* d_ws, size_t ws_size,
                              hipStream_t stream)
{
    const int*   ids  = (const int*)d_in[0];    // (8, 512) int32
    const float* attn = (const float*)d_in[1];  // (8, 64, 512) float32
    float*       out  = (float*)d_out;          // (8, 64, 50000) float32

    (void)in_sizes; (void)n_in; (void)out_size; (void)d_ws; (void)ws_size;

    dim3 grid(TDIM, BATCH, NCHUNK);             // 2048 WGs, one per (b,t,chunk)
    AttnScatter_kernel<<<grid, BLOCK, CHUNK * sizeof(float), stream>>>(ids, attn, out);
}